// RGCN_layer_20418274525632
// MI455X (gfx1250) — compile-verified
//
#include <hip/hip_runtime.h>
#include <hip/hip_bf16.h>

typedef float v2f __attribute__((ext_vector_type(2)));
typedef float v8f __attribute__((ext_vector_type(8)));

#define N_NODES 50000
#define N_EDGES 1600000
#define NUM_REL 4
#define IN_F    256
#define OUT_F   128

// ---------------------------------------------------------------------------
// Zero a u32 region (float 0.0f == 0x0 bitwise, so works for both).
// ---------------------------------------------------------------------------
__global__ void rgcn_zero_kernel(unsigned* __restrict__ p, long long n) {
    long long i = (long long)blockIdx.x * blockDim.x + threadIdx.x;
    if (i < n) p[i] = 0u;
}

// ---------------------------------------------------------------------------
// Degree counting for all 4 relations in one pass: u32 atomics (exact counts).
// edges layout: [NUM_REL][2][N_EDGES] int32.
// ---------------------------------------------------------------------------
__global__ void rgcn_degree_kernel(const int* __restrict__ edges,
                                   unsigned* __restrict__ outdeg,   // [NUM_REL][N_NODES]
                                   unsigned* __restrict__ indeg) {  // [NUM_REL][N_NODES]
    long long idx = (long long)blockIdx.x * blockDim.x + threadIdx.x;
    if (idx >= (long long)NUM_REL * N_EDGES) return;
    int r = (int)(idx / N_EDGES);
    long long e = idx - (long long)r * N_EDGES;
    int s = edges[((long long)r * 2 + 0) * N_EDGES + e];
    int d = edges[((long long)r * 2 + 1) * N_EDGES + e];
    atomicAdd(&outdeg[(long long)r * N_NODES + s], 1u);
    atomicAdd(&indeg [(long long)r * N_NODES + d], 1u);
}

// ---------------------------------------------------------------------------
// In-place count -> norm: f = rsqrt(max(count, 1))
// ---------------------------------------------------------------------------
__global__ void rgcn_norm_kernel(unsigned* __restrict__ p, long long n) {
    long long i = (long long)blockIdx.x * blockDim.x + threadIdx.x;
    if (i < n) {
        float c = (float)p[i];
        ((float*)p)[i] = rsqrtf(fmaxf(c, 1.0f));
    }
}

// ---------------------------------------------------------------------------
// H = (X * out_norm[:,None]) @ W_r  via V_WMMA_F32_16X16X4_F32 (exact fp32).
//
// Grid: N_NODES/16 = 3125 blocks (exact). Block: 256 threads = 8 waves.
//   blockIdx.x -> M tile (16 rows), wave id -> N tile (8 tiles of 16 = 128).
//
// Per ISA 16x16x4 f32 layout (wave32):
//   A (16x4): lane lid in half h holds A[M=lid][K=2h+0], A[M=lid][K=2h+1]
//   B (4x16): lane lid in half h holds B[K=2h+0][N=lid], B[K=2h+1][N=lid]
//   D (16x16): VGPR i on lane lid, half h = D[M=i+8h][N=lid]
// ---------------------------------------------------------------------------
__global__ void __launch_bounds__(256)
rgcn_gemm_wmma_kernel(const float* __restrict__ X,        // [N_NODES][IN_F]
                      const float* __restrict__ Wr,       // [IN_F][OUT_F]
                      const float* __restrict__ out_norm, // [N_NODES]
                      float* __restrict__ H) {            // [N_NODES][OUT_F]
    const int wave = threadIdx.x >> 5;       // 0..7 -> N tile
    const int lane = threadIdx.x & 31;
    const int half = lane >> 4;              // 0 or 1
    const int lid  = lane & 15;
    const int row  = blockIdx.x * 16 + lid;  // A-matrix row (M = lid)
    const int ncol = wave * 16 + lid;        // B/C/D column (N = lid)

    const float  nrm  = out_norm[row];
    const float* xrow = X + (size_t)row * IN_F;
    const float* wcol = Wr + ncol;

    v8f acc = {};
    for (int k0 = 0; k0 < IN_F; k0 += 4) {
        const int ka = k0 + 2 * half;
        v2f a = *(const v2f*)(xrow + ka);    // contiguous K pair -> b64 load
        a *= nrm;                            // fold out_norm into A
        v2f b;
        b.x = wcol[(size_t)(ka + 0) * OUT_F];
        b.y = wcol[(size_t)(ka + 1) * OUT_F];
        acc = __builtin_amdgcn_wmma_f32_16x16x4_f32(
            /*neg_a=*/false, a, /*neg_b=*/false, b,
            /*c_mod=*/(short)0, acc, /*reuse_a=*/false, /*reuse_b=*/false);
    }

#pragma unroll
    for (int i = 0; i < 8; ++i) {
        const int m = blockIdx.x * 16 + i + 8 * half;
        H[(size_t)m * OUT_F + ncol] = acc[i];
    }
}

// ---------------------------------------------------------------------------
// Edge scatter: Z[dst] += in_norm[dst] * H[src]
// 32 lanes per edge, 4 features per lane: b128 gather + 4 f32 atomics.
// ---------------------------------------------------------------------------
__global__ void __launch_bounds__(256)
rgcn_scatter_kernel(const float* __restrict__ H,        // [N_NODES][OUT_F]
                    const int* __restrict__ src,        // [N_EDGES]
                    const int* __restrict__ dst,        // [N_EDGES]
                    const float* __restrict__ in_norm,  // [N_NODES]
                    float* __restrict__ Z) {            // [N_NODES][OUT_F]
    long long tid = (long long)blockIdx.x * blockDim.x + threadIdx.x;
    long long e = tid >> 5;
    int lane = (int)(tid & 31);
    if (e >= N_EDGES) return;

    const int s = src[e];
    const int d = dst[e];
    const float nrm = in_norm[d];

    const float4 h4 = *(const float4*)(H + (size_t)s * OUT_F + lane * 4);
    float* zp = Z + (size_t)d * OUT_F + lane * 4;
    atomicAdd(zp + 0, h4.x * nrm);
    atomicAdd(zp + 1, h4.y * nrm);
    atomicAdd(zp + 2, h4.z * nrm);
    atomicAdd(zp + 3, h4.w * nrm);
}

// ---------------------------------------------------------------------------
// Launch
// ---------------------------------------------------------------------------
extern "C" void kernel_launch(void* const* d_in, const int* in_sizes, int n_in,
                              void* d_out, int out_size, void* d_ws, size_t ws_size,
                              hipStream_t stream) {
    const float* X     = (const float*)d_in[0];   // [N_NODES][IN_F]
    const float* W     = (const float*)d_in[1];   // [NUM_REL][IN_F][OUT_F]
    const int*   edges = (const int*)d_in[2];     // [NUM_REL][2][N_EDGES]
    float*       Z     = (float*)d_out;           // [N_NODES][OUT_F]

    // Workspace layout (u32/float):
    //   outdeg : NUM_REL * N_NODES u32   (becomes out_norm after norm pass)
    //   indeg  : NUM_REL * N_NODES u32   (becomes in_norm)
    //   H      : N_NODES * OUT_F float   (per-relation GEMM result, reused)
    unsigned* outdeg = (unsigned*)d_ws;
    unsigned* indeg  = outdeg + (size_t)NUM_REL * N_NODES;
    float*    H      = (float*)(indeg + (size_t)NUM_REL * N_NODES);

    const int T = 256;

    // Zero output accumulator and degree scratch.
    {
        long long nz = (long long)N_NODES * OUT_F;
        rgcn_zero_kernel<<<(unsigned)((nz + T - 1) / T), T, 0, stream>>>((unsigned*)Z, nz);
        long long nd = 2LL * NUM_REL * N_NODES;
        rgcn_zero_kernel<<<(unsigned)((nd + T - 1) / T), T, 0, stream>>>(outdeg, nd);
    }

    // Degrees for all relations, then counts -> rsqrt norms in place.
    {
        long long ne = (long long)NUM_REL * N_EDGES;
        rgcn_degree_kernel<<<(unsigned)((ne + T - 1) / T), T, 0, stream>>>(edges, outdeg, indeg);
        long long nd = 2LL * NUM_REL * N_NODES;
        rgcn_norm_kernel<<<(unsigned)((nd + T - 1) / T), T, 0, stream>>>(outdeg, nd);
    }

    // Per relation: WMMA GEMM into H, then scatter-add into Z.
    for (int r = 0; r < NUM_REL; ++r) {
        const float* Wr       = W + (size_t)r * IN_F * OUT_F;
        const float* out_norm = (const float*)(outdeg + (size_t)r * N_NODES);
        const float* in_norm  = (const float*)(indeg  + (size_t)r * N_NODES);
        const int*   srcs     = edges + ((size_t)r * 2 + 0) * N_EDGES;
        const int*   dsts     = edges + ((size_t)r * 2 + 1) * N_EDGES;

        rgcn_gemm_wmma_kernel<<<N_NODES / 16, T, 0, stream>>>(X, Wr, out_norm, H);

        long long nt = (long long)N_EDGES * 32;
        rgcn_scatter_kernel<<<(unsigned)((nt + T - 1) / T), T, 0, stream>>>(
            H, srcs, dsts, in_norm, Z);
    }
}